// KANN_11055245820078
// MI455X (gfx1250) — compile-verified
//
#include <hip/hip_runtime.h>
#include <stdint.h>

// ---- problem constants (from reference) ----
#define NSAMP   32768
#define NW      32          // widths K
#define NDIM    4
#define NNODES  193
#define NELEM   64
#define SPB     128         // samples per block
#define TPB     256         // 8 waves (wave32)

#define THIRD   (1.0f / 3.0f)   // avoid IEEE div sequence; only feeds floor+clamp

// ---- dynamic LDS layout (bytes) ----
#define OFF_WIE      0                   // 131072 B : W_in  element layout [k][j][el][p] f32
#define OFF_WOE      131072              // 32768  B : W_out element layout [k][el][p]    f32
#define OFF_C        163840              // union region
#define OFF_STG_WIN  (OFF_C)             // 98816 B raw W_in staging (phase 0 only)
#define OFF_STG_WOUT (OFF_C + 98816)     // 24704 B raw W_out staging (phase 0 only)
#define OFF_BASIS    (OFF_C)             // 512*16*4 = 32768 B basis table (phase 1+)
#define OFF_ELT      (OFF_C + 32768)     // 512*4  = 2048 B element ids
#define LDS_BYTES    (OFF_C + 98816 + 24704)   // 287360 <= 320KB/WGP

typedef __attribute__((ext_vector_type(2))) float v2f;
typedef __attribute__((ext_vector_type(8))) float v8f;

// Power-basis coefficients of the 12 basis functions (cols): n=0..3 phi_p,
// 4..7 dphi_p, 8..11 ddphi_p, 12..15 zero pad.  Row = power of x (0..3).
// Nodes are [-1,-1/3,1/3,1].
__constant__ float BCOEF[4][16] = {
  { -0.0625f,  0.5625f,  0.5625f, -0.0625f,
     0.0625f, -1.6875f,  1.6875f, -0.0625f,
     1.125f,  -1.125f,  -1.125f,   1.125f,   0.f, 0.f, 0.f, 0.f },
  {  0.0625f, -1.6875f,  1.6875f, -0.0625f,
     1.125f,  -1.125f,  -1.125f,   1.125f,
    -3.375f,  10.125f, -10.125f,   3.375f,   0.f, 0.f, 0.f, 0.f },
  {  0.5625f, -0.5625f, -0.5625f,  0.5625f,
    -1.6875f,  5.0625f, -5.0625f,  1.6875f,
     0.f, 0.f, 0.f, 0.f,                     0.f, 0.f, 0.f, 0.f },
  { -0.5625f,  1.6875f, -1.6875f,  0.5625f,
     0.f, 0.f, 0.f, 0.f,  0.f, 0.f, 0.f, 0.f, 0.f, 0.f, 0.f, 0.f }
};

// ---- CDNA5 async global->LDS copy (16B per lane), ASYNCcnt-tracked ----
#if __has_builtin(__builtin_amdgcn_global_load_async_to_lds_b128)
typedef int v4i_vs __attribute__((vector_size(16)));
typedef __attribute__((address_space(1))) v4i_vs* as1_v4i;
typedef __attribute__((address_space(3))) v4i_vs* as3_v4i;
__device__ __forceinline__ void async_g2l_b128(const void* gsrc, void* lds_dst) {
  __builtin_amdgcn_global_load_async_to_lds_b128(
      (as1_v4i)gsrc, (as3_v4i)lds_dst, 0, 0);
}
#else
__device__ __forceinline__ void async_g2l_b128(const void* gsrc, void* lds_dst) {
  unsigned lds_off = (unsigned)(unsigned long long)lds_dst; // flat[31:0] == LDS offset
  asm volatile("global_load_async_to_lds_b128 %0, %1, off"
               :: "v"(lds_off), "v"(gsrc) : "memory");
}
#endif

__device__ __forceinline__ void wait_async0() {
#if __has_builtin(__builtin_amdgcn_s_wait_asynccnt)
  __builtin_amdgcn_s_wait_asynccnt(0);
#else
  asm volatile("s_wait_asynccnt 0" ::: "memory");
#endif
}

extern "C" __global__ void __launch_bounds__(TPB)
kann_fused(const float* __restrict__ x, const float* __restrict__ Win,
           const float* __restrict__ Wout, float* __restrict__ out)
{
  extern __shared__ char smem[];
  const unsigned tid = threadIdx.x;
  const unsigned blk = blockIdx.x;

  // ================= phase 0a: async-stage raw weights into LDS ===========
  // W_in: 32*193 nodes * 16B, W_out: 1544 * 16B  (both 16B aligned)
  for (unsigned u = tid; u < (6176u + 1544u); u += TPB) {
    if (u < 6176u)
      async_g2l_b128(Win + (size_t)u * 4, smem + OFF_STG_WIN + (size_t)u * 16);
    else
      async_g2l_b128(Wout + (size_t)(u - 6176u) * 4,
                     smem + OFF_STG_WOUT + (size_t)(u - 6176u) * 16);
  }
  wait_async0();
  __syncthreads();

  // ====== phase 0b: transpose to element-replicated gather layout =========
  // W_in_e[k][j][el][p] = W_in[k][3*el+p][j]  (node 3e shared with prev elem p=3)
  {
    float* WIE = (float*)(smem + OFF_WIE);
    float* WOE = (float*)(smem + OFF_WOE);
    for (unsigned u = tid; u < 32u * 193u; u += TPB) {
      unsigned k = u / 193u, n = u % 193u;
      float4 w = *(const float4*)(smem + OFF_STG_WIN + (size_t)u * 16);
      float wo = *(const float*)(smem + OFF_STG_WOUT + (size_t)u * 4);
      unsigned e = n / 3u, p = n % 3u;
      if (e < 64u) {
#pragma unroll
        for (int j = 0; j < 4; ++j)
          WIE[(((k * 4u + j) * 64u + e) * 4u) + p] = ((const float*)&w)[j];
        WOE[(k * 64u + e) * 4u + p] = wo;
      }
      if (p == 0u && n > 0u) {          // this node is also p=3 of previous elem
        unsigned e0 = e - 1u;
#pragma unroll
        for (int j = 0; j < 4; ++j)
          WIE[(((k * 4u + j) * 64u + e0) * 4u) + 3u] = ((const float*)&w)[j];
        WOE[(k * 64u + e0) * 4u + 3u] = wo;
      }
    }
  }
  __syncthreads();

  // ============ phase 1: basis values for 512 (sample,dim) pairs ===========
  // [1,x,x^2,x^3](16 pairs) @ BCOEF(4x16)  ->  12 basis values per pair
  {
    const unsigned lane = tid & 31u;
    const unsigned wv   = tid >> 5;       // wave id 0..7
    const unsigned lo   = lane & 15u;
    const bool upper    = lane >= 16u;    // K-half split per 32-bit A layout
    float* bt  = (float*)(smem + OFF_BASIS);
    int*   elt = (int*)(smem + OFF_ELT);
#if __has_builtin(__builtin_amdgcn_wmma_f32_16x16x4_f32)
    v2f b;                                // B: K rows split across lane halves
    b.x = BCOEF[upper ? 2 : 0][lo];
    b.y = BCOEF[upper ? 3 : 1][lo];
#pragma unroll
    for (unsigned it = 0; it < 4; ++it) {
      unsigned pbase = wv * 64u + it * 16u;
      unsigned pair  = pbase + lo;        // lanes 16..31 mirror 0..15
      float xv  = x[(size_t)blk * (SPB * 4) + pair];
      float xs  = 96.0f * (xv + 1.0f);                     // layer_transform, inner
      float elf = fminf(fmaxf(floorf(xs * THIRD), 0.0f), 63.0f);
      float xr  = (xs - 3.0f * elf) * (2.0f / 3.0f) - 1.0f;
      if (!upper) elt[pair] = (int)elf;
      float xr2 = xr * xr, xr3 = xr2 * xr;
      v2f a; a.x = upper ? xr2 : 1.0f; a.y = upper ? xr3 : xr;
      v8f c = {};
      v8f d = __builtin_amdgcn_wmma_f32_16x16x4_f32(
          false, a, false, b, (short)0, c, false, false);
      unsigned moff = upper ? 8u : 0u;    // D rows: lanes<16 -> r, lanes>=16 -> r+8
#pragma unroll
      for (int r = 0; r < 8; ++r)
        bt[(pbase + moff + (unsigned)r) * 16u + lo] = d[r];
    }
#else
    for (unsigned pair = tid; pair < SPB * 4u; pair += TPB) {
      float xv  = x[(size_t)blk * (SPB * 4) + pair];
      float xs  = 96.0f * (xv + 1.0f);
      float elf = fminf(fmaxf(floorf(xs * THIRD), 0.0f), 63.0f);
      float xr  = (xs - 3.0f * elf) * (2.0f / 3.0f) - 1.0f;
      elt[pair] = (int)elf;
#pragma unroll
      for (int n = 0; n < 12; ++n)
        bt[pair * 16u + n] =
            ((BCOEF[3][n] * xr + BCOEF[2][n]) * xr + BCOEF[1][n]) * xr + BCOEF[0][n];
    }
#endif
  }
  __syncthreads();

  // ====== phase 2: per-(sample, width) accumulation + outer layer ==========
  {
    const unsigned k = tid & 31u;                // lane == width -> coalesced stores
    const float* WIE = (const float*)(smem + OFF_WIE);
    const float* WOE = (const float*)(smem + OFF_WOE);
    const float* BT  = (const float*)(smem + OFF_BASIS);
    const int*   ELT = (const int*)(smem + OFF_ELT);
    const size_t IK  = (size_t)NSAMP * NW;
#pragma unroll 1
    for (unsigned it = 0; it < 16; ++it) {
      unsigned s = it * 8u + (tid >> 5);
      unsigned i = blk * SPB + s;
      int4 el4 = *(const int4*)(ELT + s * 4u);
      float t = 0.f, dtr = 0.f, ddtr = 0.f;
#pragma unroll
      for (int j = 0; j < 4; ++j) {
        const float* bp = BT + (s * 4u + (unsigned)j) * 16u;
        float4 ph = *(const float4*)(bp);
        float4 dp = *(const float4*)(bp + 4);
        float4 d2 = *(const float4*)(bp + 8);
        int el = (&el4.x)[j];
        float4 wv = *(const float4*)(WIE + (((k * 4u + (unsigned)j) * 64u +
                                             (unsigned)el) * 4u));
        t    += wv.x * ph.x + wv.y * ph.y + wv.z * ph.z + wv.w * ph.w;
        dtr  += wv.x * dp.x + wv.y * dp.y + wv.z * dp.z + wv.w * dp.w;
        ddtr += wv.x * d2.x + wv.y * d2.y + wv.z * d2.z + wv.w * d2.w;
      }
      float dt  = dtr * 64.0f;       // 1/delta,  delta = 3*2/(2*192) = 1/64
      float ddt = ddtr * 4096.0f;    // 1/delta^2

      // outer layer on t over [-3,3]
      float xs2 = (t + 3.0f) * 32.0f;
      float e2f = fminf(fmaxf(floorf(xs2 * THIRD), 0.0f), 63.0f);
      float xr  = (xs2 - 3.0f * e2f) * (2.0f / 3.0f) - 1.0f;
      float4 w2 = *(const float4*)(WOE + ((k * 64u + (unsigned)(int)e2f) * 4u));
      float y = 0.f;
#pragma unroll
      for (int p = 0; p < 4; ++p) {
        float cc = ((BCOEF[3][p] * xr + BCOEF[2][p]) * xr + BCOEF[1][p]) * xr
                   + BCOEF[0][p];
        y += (&w2.x)[p] * cc;
      }
      size_t base = (size_t)i * NW + k;
      out[0 * IK + base] = y;
      out[1 * IK + base] = t;
      out[2 * IK + base] = dt;
      out[3 * IK + base] = ddt;
    }
  }
}

extern "C" void kernel_launch(void* const* d_in, const int* in_sizes, int n_in,
                              void* d_out, int out_size, void* d_ws, size_t ws_size,
                              hipStream_t stream) {
  const float* x    = (const float*)d_in[0];
  const float* Win  = (const float*)d_in[1];
  const float* Wout = (const float*)d_in[2];
  float* out = (float*)d_out;
  (void)in_sizes; (void)n_in; (void)out_size; (void)d_ws; (void)ws_size;
  kann_fused<<<dim3(NSAMP / SPB), dim3(TPB), LDS_BYTES, stream>>>(x, Win, Wout, out);
}